// CTCDecoder_57681410785543
// MI455X (gfx1250) — compile-verified
//
#include <hip/hip_runtime.h>
#include <cstdint>
#include <cstddef>

// Problem shape (fixed by the reference's setup_inputs)
#define BB 16
#define TT 800
#define VV 5000
#define LL 100
#define SS 201           // 2*L + 1
#define NEG_INF (-1.0e30f)

typedef __attribute__((ext_vector_type(2))) float v2f;
typedef __attribute__((ext_vector_type(8))) float v8f;

// ---------------------------------------------------------------------------
// Wave32 sum-reduction via V_WMMA_F32_16X16X4_F32 with B = ones:
// a = {x, 0} puts lane L's partial at A[m= L%16, k = (L<16)?0:2], so
// rowsum(m) = x_m + x_{m+16}; D[m,n] = rowsum(m) for every n. Summing the 8
// D VGPRs per lane gives rows 0-7 (lanes 0-15) / rows 8-15 (lanes 16-31);
// one xor-16 fold yields the full 32-lane sum in every lane.
// ---------------------------------------------------------------------------
__device__ __forceinline__ float wave_sum_wmma(float x) {
    v2f a; a.x = x;    a.y = 0.0f;
    v2f b; b.x = 1.0f; b.y = 1.0f;
    v8f c = {};
    v8f d = __builtin_amdgcn_wmma_f32_16x16x4_f32(
        /*neg_a=*/false, a, /*neg_b=*/false, b,
        /*c_mod=*/(short)0, c, /*reuse_a=*/false, /*reuse_b=*/false);
    float s = d[0] + d[1] + d[2] + d[3] + d[4] + d[5] + d[6] + d[7];
    s += __shfl_xor(s, 16, 32);
    return s;
}

// ---------------------------------------------------------------------------
// Kernel 1: one block per (b,t) row. Async-copy the 20KB logits row into LDS
// (non-temporal: logits are read exactly once; keep lp L2-resident), then
// logsumexp-reduce and gather the S=201 extended-symbol log-probs.
// Global traffic: 256MB read once (roofline ~11us at 23.3 TB/s).
// ---------------------------------------------------------------------------
__global__ __launch_bounds__(256)
void ctc_lse_gather(const float* __restrict__ logits,
                    const int*   __restrict__ targets,
                    float*       __restrict__ lp) {
    __shared__ float smem[VV];
    __shared__ float redm[8];
    __shared__ float reds[8];

    const int row  = blockIdx.x;        // row = b*TT + t
    const int b    = row / TT;
    const int tid  = threadIdx.x;
    const int lane = tid & 31;
    const int wid  = tid >> 5;

    // ---- CDNA5 async global->LDS copy of one row (1250 x 16B chunks) ----
    const uint64_t gbase = (uint64_t)(uintptr_t)logits + (uint64_t)row * (VV * 4);
    const uint32_t lbase = (uint32_t)(uintptr_t)(void*)smem; // flat low32 == LDS addr
    const int nchunk = (VV * 4) / 16;   // 1250, exact
    for (int k = tid; k < nchunk; k += 256) {
        uint32_t la = lbase + (uint32_t)k * 16u;
        uint64_t ga = gbase + (uint64_t)k * 16u;
        asm volatile("global_load_async_to_lds_b128 %0, %1, off th:TH_LOAD_NT"
                     :: "v"(la), "v"(ga) : "memory");
    }
    asm volatile("s_wait_asynccnt 0x0" ::: "memory");
    __syncthreads();

    // ---- row max ----
    float mx = -3.0e38f;
    for (int i = tid; i < VV; i += 256) mx = fmaxf(mx, smem[i]);
#pragma unroll
    for (int off = 16; off; off >>= 1) mx = fmaxf(mx, __shfl_xor(mx, off, 32));
    if (lane == 0) redm[wid] = mx;
    __syncthreads();
    float rowmax = redm[0];
#pragma unroll
    for (int w = 1; w < 8; ++w) rowmax = fmaxf(rowmax, redm[w]);

    // ---- sum of exp (wave reduction through the WMMA pipe) ----
    float acc = 0.0f;
    for (int i = tid; i < VV; i += 256) acc += __expf(smem[i] - rowmax);
    float wsum = wave_sum_wmma(acc);
    if (lane == 0) reds[wid] = wsum;
    __syncthreads();
    float tot = reds[0] + reds[1] + reds[2] + reds[3] +
                reds[4] + reds[5] + reds[6] + reds[7];
    const float lse = rowmax + __logf(tot);   // tot in [1, 5000]

    // ---- gather extended-symbol log-probs: ext[s] = blank(0) or targets ----
    if (tid < SS) {
        int sym = (tid & 1) ? targets[b * LL + (tid >> 1)] : 0;
        lp[(size_t)row * SS + tid] = smem[sym] - lse;
    }
}

// ---------------------------------------------------------------------------
// Kernel 2: alpha recursion. One wave32 per batch element. CONTIGUOUS state
// layout: lane L owns s = 7L + j (j=0..6), so alpha[s-1]/alpha[s-2] are
// in-lane registers for j>=2; only the lane boundary needs cross-lane data:
// exactly 2 shfl_up per timestep (prev lane's alpha[6] and alpha[5]).
// 3-way log-sum-exp keeps the dependent chain short; lp rows are
// double-buffered (L2-resident) to hide load latency across the 800 steps.
// ---------------------------------------------------------------------------
__device__ __forceinline__ void load_row(float dst[7], const float* __restrict__ lp,
                                         int b, int t, int lane) {
    const float* r = lp + (size_t)(b * TT + t) * SS;
#pragma unroll
    for (int j = 0; j < 7; ++j) {
        int s = 7 * lane + j;
        dst[j] = (s < SS) ? r[s] : NEG_INF;
    }
}

__global__ __launch_bounds__(32)
void ctc_alpha(const float* __restrict__ lp,
               const int*   __restrict__ targets,
               const int*   __restrict__ in_len,
               const int*   __restrict__ tgt_len,
               float*       __restrict__ out) {
    const int b    = blockIdx.x;
    const int lane = threadIdx.x;
    const int len  = in_len[b];               // in [T/2, T]
    const int* tg  = targets + b * LL;
    __shared__ float fin[224];

    // allow_skip[s]: s odd, s>=3, ext[s] != ext[s-2]
    bool skipok[7];
#pragma unroll
    for (int j = 0; j < 7; ++j) {
        int s = 7 * lane + j;
        bool ok = (s & 1) && (s >= 3) && (s < SS);
        int c = ok ? tg[(s - 1) >> 1] : 0;
        int p = ok ? tg[(s - 3) >> 1] : 1;
        skipok[j] = ok && (c != p);
    }

    // alpha(t=0): only s=0,1 are live (both in lane 0)
    const float* row0 = lp + (size_t)(b * TT) * SS;
    float alpha[7];
#pragma unroll
    for (int j = 0; j < 7; ++j) {
        int s = 7 * lane + j;
        float v = NEG_INF;
        if (s == 0) v = row0[0];
        if (s == 1) v = row0[1];
        alpha[j] = v;
    }

    float cur[7];
    load_row(cur, lp, b, 1, lane);            // len >= 400 > 1 always

    for (int t = 1; t < len; ++t) {
        float nxt[7];
        if (t + 1 < len) load_row(nxt, lp, b, t + 1, lane);
        else {
#pragma unroll
            for (int j = 0; j < 7; ++j) nxt[j] = NEG_INF;
        }

        // cross-lane boundary values (old alpha of the previous lane)
        float p6 = __shfl_up(alpha[6], 1, 32); // alpha[7*(L-1)+6] = alpha[s0-1]
        float p5 = __shfl_up(alpha[5], 1, 32); // alpha[7*(L-1)+5] = alpha[s0-2]
        if (lane == 0) { p6 = NEG_INF; p5 = NEG_INF; }

        float anew[7];
#pragma unroll
        for (int j = 0; j < 7; ++j) {
            float aj = alpha[j];
            float a1 = (j == 0) ? p6 : alpha[j - 1];
            float a2 = (j == 0) ? p5 : ((j == 1) ? p6 : alpha[j - 2]);
            a2 = skipok[j] ? a2 : NEG_INF;
            // 3-way log-sum-exp (== sequential logaddexp in exact math)
            float m   = fmaxf(fmaxf(aj, a1), a2);
            float sum = __expf(aj - m) + __expf(a1 - m) + __expf(a2 - m);
            anew[j] = m + __logf(sum) + cur[j];
        }
#pragma unroll
        for (int j = 0; j < 7; ++j) { alpha[j] = anew[j]; cur[j] = nxt[j]; }
    }

#pragma unroll
    for (int j = 0; j < 7; ++j) fin[7 * lane + j] = alpha[j];
    __syncthreads();
    if (lane == 0) {
        int sl = 2 * tgt_len[b];              // in [100, 200]
        float x = fin[sl], y = fin[sl - 1];
        float m = fmaxf(x, y);
        out[b] = -(m + __logf(__expf(x - m) + __expf(y - m)));
    }
}

// ---------------------------------------------------------------------------
extern "C" void kernel_launch(void* const* d_in, const int* in_sizes, int n_in,
                              void* d_out, int out_size, void* d_ws, size_t ws_size,
                              hipStream_t stream) {
    (void)in_sizes; (void)n_in; (void)out_size; (void)ws_size;
    const float* logits  = (const float*)d_in[0];
    const int*   targets = (const int*)d_in[1];
    const int*   in_len  = (const int*)d_in[2];
    const int*   tgt_len = (const int*)d_in[3];
    float*       out     = (float*)d_out;
    float*       lp      = (float*)d_ws;      // B*T*S fp32 = ~10.3 MB scratch

    ctc_lse_gather<<<BB * TT, 256, 0, stream>>>(logits, targets, lp);
    ctc_alpha<<<BB, 32, 0, stream>>>(lp, targets, in_len, tgt_len, out);
}